// DiscreteBayesianFlow_70669391888455
// MI455X (gfx1250) — compile-verified
//
#include <hip/hip_runtime.h>

// CDNA5 / gfx1250: wave32, WMMA F32 16x16x4 for the triangular noise @ L0^T
// GEMM feeding softmax. L0^T = strictlyUpper(e_k) + diag(d_k) (closed-form
// equicorrelated Cholesky). WMMA handles the strictly-upper part: interior
// blocks need zero compares, the 4 diagonal-crossing steps per tile use
// multiply-by-{0,1} masks (un-sinkable loads -> pure cndmask, no exec
// diamonds). The diag(d) term is a single FMA in the epilogue. Zero blocks
// skipped; waves get balanced tile sets (equal WMMA counts).

typedef __attribute__((ext_vector_type(2))) float v2f;
typedef __attribute__((ext_vector_type(8))) float v8f;

#define KCLS 256
#define ROWS_PER_BLK 16
#define LDS_STRIDE 260          // 256 + 4 pad; 260*4 B is 16B aligned; banks 4m+k -> conflict-free
#define NTHREADS 128            // 4 wave32

__global__ __launch_bounds__(NTHREADS)
void bfn_probs_kernel(const int* __restrict__ data,
                      const float* __restrict__ tin,
                      const float* __restrict__ noise,
                      float* __restrict__ out)
{
    __shared__ __align__(16) float Abuf[ROWS_PER_BLK * LDS_STRIDE]; // noise tile, later logits/exp
    __shared__ __align__(8) float dtab[KCLS];   // Cholesky diagonal d_k (epilogue only)
    __shared__ __align__(8) float etab[KCLS];   // Cholesky sub-diagonal value e_k
    __shared__ float sbS[ROWS_PER_BLK], betaS[ROWS_PER_BLK], invS[ROWS_PER_BLK];
    __shared__ int   clsS[ROWS_PER_BLK], loS[ROWS_PER_BLK];
    __shared__ float red1[ROWS_PER_BLK][8];
    __shared__ float red2[ROWS_PER_BLK][8];

    const int tid  = threadIdx.x;
    const int row0 = blockIdx.x * ROWS_PER_BLK;

    // ---- Stage 1a: closed-form Cholesky of C0 = v*I - J (v = 256.001):
    //      b_k = v/(k - v); d_k = sqrt(v + b_k); e_k = b_k/d_k.
    for (int k = tid; k < KCLS; k += NTHREADS) {
        const float v = 256.001f;
        float b = v / ((float)k - v);
        float d = sqrtf(v + b);
        dtab[k] = d;
        etab[k] = b / d;
    }

    // ---- Stage 1b: per-row scalars (beta schedule, class index)
    if (tid < ROWS_PER_BLK) {
        float tv  = tin[row0 + tid];
        float sb0 = fminf(tv, 1.0f - 1e-6f);   // clip(t, None, 1-eps) * MAX_SQRT_BETA(=1)
        loS[tid]  = (sb0 < 1e-10f) ? 1 : 0;
        float sb  = fmaxf(sb0, 1e-10f);
        sbS[tid]   = sb;                        // sqrt(beta)
        betaS[tid] = sb * sb;                   // beta
        clsS[tid]  = data[row0 + tid];
    }

    // ---- Stage 1c: noise tile [16 rows x 256 k] -> LDS (float4 coalesced)
    {
        const float4* src = (const float4*)(noise + (size_t)row0 * KCLS);
        #pragma unroll
        for (int i = 0; i < 8; ++i) {
            int f   = tid + i * NTHREADS;       // float4 index 0..1023
            int row = f >> 6;
            int kq  = (f & 63) << 2;
            float4 val = src[f];
            *(float4*)&Abuf[row * LDS_STRIDE + kq] = val;
        }
    }
    __syncthreads();

    // ---- Stage 2: strictly-upper triangular GEMM
    //      acc[m][n] = sum_{k<n} noise[m][k] * e_k     (diag added in epilogue)
    const int lane = tid & 31;
    // Wave id pinned to SGPR: tile ids / loop bounds scalar, EXEC untouched.
    const int wave = __builtin_amdgcn_readfirstlane(tid >> 5);
    const int half = lane >> 4;          // A/B frag: lanes 0-15 hold K pair {0,1}, 16-31 {2,3}
    const int nl   = lane & 15;
    const int m    = nl;                 // A-matrix row for this lane
    const int koff = half * 2;

    // Balanced tile assignment: tile cost ~ n0, sets {w, 7-w, 8+w, 15-w} all sum equal.
    const int tlist[4] = { wave, 7 - wave, 8 + wave, 15 - wave };
    v8f acc[4];

    #pragma unroll
    for (int j = 0; j < 4; ++j) {
        const int n0 = tlist[j] * 16;    // scalar (SGPR)
        const int n  = n0 + nl;          // this lane's output column (vector)
        v8f c = {};

        // Interior region: all k < n0 <= n  =>  B is pure e_k (no compares).
        #pragma unroll 2
        for (int kk = 0; kk < n0; kk += 4) {
            const int k0 = kk + koff;
            v2f a, b;
            *(float2*)&a = *(const float2*)&Abuf[m * LDS_STRIDE + k0];
            *(float2*)&b = *(const float2*)&etab[k0];      // broadcast within half-wave
            c = __builtin_amdgcn_wmma_f32_16x16x4_f32(
                    false, a, false, b, (short)0, c, false, false);
        }

        // Diagonal-crossing region: kk in [n0, n0+16). Strictly-upper only:
        // b = e_k * mask(n > k). mask is a select of literals -> the e-load is
        // used unconditionally, so it cannot be sunk into a branch.
        #pragma unroll
        for (int s = 0; s < 4; ++s) {
            const int k0 = n0 + s * 4 + koff;
            v2f a, b;
            *(float2*)&a = *(const float2*)&Abuf[m * LDS_STRIDE + k0];
            const float2 ep = *(const float2*)&etab[k0];
            b[0] = ep.x * ((n > k0    ) ? 1.0f : 0.0f);
            b[1] = ep.y * ((n > k0 + 1) ? 1.0f : 0.0f);
            c = __builtin_amdgcn_wmma_f32_16x16x4_f32(
                    false, a, false, b, (short)0, c, false, false);
        }
        // k >= n0+16: block is all-zero -> skipped entirely.
        acc[j] = c;
    }

    __syncthreads();   // all waves done reading Abuf noise for WMMA

    // ---- Stage 3: y = acc + d_n*noise[m][n];  logits = sqrt(beta)*y + beta*(K*onehot-1)
    // Each lane reads its own (row,n) noise slot then overwrites it with the
    // logit: same-thread read-then-write, exclusive ownership, no race.
    #pragma unroll
    for (int j = 0; j < 4; ++j) {
        const int n  = tlist[j] * 16 + nl;
        const float dn = dtab[n];
        #pragma unroll
        for (int r = 0; r < 8; ++r) {
            const int row = r + half * 8;   // D layout: VGPR r = row r (lanes 0-15) / r+8
            float* slot = &Abuf[row * LDS_STRIDE + n];
            float y = fmaf(dn, *slot, acc[j][r]);       // add diagonal term
            float logit = sbS[row] * y - betaS[row]
                        + (clsS[row] == n ? 256.0f * betaS[row] : 0.0f);
            *slot = logit;
        }
    }
    __syncthreads();

    // ---- Stage 4: row softmax (8 threads per row, 32 cols each)
    {
        const int srow = tid >> 3;
        const int seg  = tid & 7;
        float* rowp = &Abuf[srow * LDS_STRIDE + seg * 32];
        float mx = -3.4028235e38f;
        #pragma unroll
        for (int c = 0; c < 32; ++c) mx = fmaxf(mx, rowp[c]);
        red1[srow][seg] = mx;
        __syncthreads();
        float rmx = red1[srow][0];
        #pragma unroll
        for (int c = 1; c < 8; ++c) rmx = fmaxf(rmx, red1[srow][c]);
        float s = 0.0f;
        #pragma unroll
        for (int c = 0; c < 32; ++c) {
            float e = __expf(rowp[c] - rmx);
            rowp[c] = e;                 // stash exp for final pass
            s += e;
        }
        red2[srow][seg] = s;
        __syncthreads();
        if (seg == 0) {
            float tot = 0.0f;
            #pragma unroll
            for (int c = 0; c < 8; ++c) tot += red2[srow][c];
            invS[srow] = 1.0f / tot;
        }
    }
    __syncthreads();

    // ---- Stage 5: normalize + lo_beta override, coalesced store
    // (load hoisted out of the ternary so the select is register-register)
    float* dst = out + (size_t)row0 * KCLS;
    #pragma unroll
    for (int i = 0; i < 32; ++i) {
        int flat = tid + i * NTHREADS;       // 0..4095
        int row  = flat >> 8;
        int col  = flat & 255;
        float p   = Abuf[row * LDS_STRIDE + col] * invS[row];
        float val = loS[row] ? (1.0f / 256.0f) : p;
        dst[flat] = val;
    }
}

extern "C" void kernel_launch(void* const* d_in, const int* in_sizes, int n_in,
                              void* d_out, int out_size, void* d_ws, size_t ws_size,
                              hipStream_t stream) {
    const int*   data  = (const int*)  d_in[0];   // [B,S] int32 class indices
    const float* tin   = (const float*)d_in[1];   // [B,S] fp32
    const float* noise = (const float*)d_in[2];   // [B,S,256] fp32
    float* out = (float*)d_out;                   // [B,S,256] fp32 probs

    const int rows   = in_sizes[0];               // B*S = 16384
    const int blocks = rows / ROWS_PER_BLK;       // 1024
    bfn_probs_kernel<<<blocks, NTHREADS, 0, stream>>>(data, tin, noise, out);
}